// RungeKuttaIntegratorCell__9466107921018
// MI455X (gfx1250) — compile-verified
//
#include <hip/hip_runtime.h>
#include <hip/hip_bf16.h>

typedef __attribute__((ext_vector_type(2))) float v2f;
typedef __attribute__((ext_vector_type(8))) float v8f;

#define RK_DT 0.001f
#define INV_95_45 (1.0f / 95.45f)   // constant-folded reciprocal of the divisor

// Single-state RBF dot via V_WMMA_F32_16X16X4_F32.
// A (16x4) = weights replicated over rows; B (4x16) = phi, N = element.
// D rows all identical => every lane reads its own element's dot from D vgpr0.
template <int NCH>
__device__ __forceinline__ float rbf_wmma(float y, int k0,
                                          const float* __restrict__ sC,
                                          const float* __restrict__ sG,
                                          const float* __restrict__ sW,
                                          int nch_rt) {
    const int nch = (NCH > 0) ? NCH : nch_rt;
    v8f acc = {0.f, 0.f, 0.f, 0.f, 0.f, 0.f, 0.f, 0.f};
#pragma unroll 4
    for (int ch = 0; ch < nch; ++ch) {
        const int k = (ch << 2) + k0;
        const float c0 = sC[k], c1 = sC[k + 1];
        const float g0 = sG[k], g1 = sG[k + 1];
        v2f a;                                // weights (same for all M rows)
        a.x = sW[k];
        a.y = sW[k + 1];
        const float t0 = (y - c0) * g0;
        const float t1 = (y - c1) * g1;
        v2f b;                                // phi for this lane's element
        b.x = __expf(-(t0 * t0));
        b.y = __expf(-(t1 * t1));
        acc = __builtin_amdgcn_wmma_f32_16x16x4_f32(
            false, a, false, b, (short)0, acc, false, false);
    }
    return acc[0];
}

// Fused two-state RBF: shares LDS table loads and the A operand between the
// k2/k3 state (ya) and the k4 state (yb); two independent WMMA accumulators.
template <int NCH>
__device__ __forceinline__ void rbf_wmma_dual(float ya, float yb, int k0,
                                              const float* __restrict__ sC,
                                              const float* __restrict__ sG,
                                              const float* __restrict__ sW,
                                              int nch_rt,
                                              float& ra, float& rb) {
    const int nch = (NCH > 0) ? NCH : nch_rt;
    v8f acca = {0.f, 0.f, 0.f, 0.f, 0.f, 0.f, 0.f, 0.f};
    v8f accb = {0.f, 0.f, 0.f, 0.f, 0.f, 0.f, 0.f, 0.f};
#pragma unroll 4
    for (int ch = 0; ch < nch; ++ch) {
        const int k = (ch << 2) + k0;
        const float c0 = sC[k], c1 = sC[k + 1];
        const float g0 = sG[k], g1 = sG[k + 1];
        v2f a;
        a.x = sW[k];
        a.y = sW[k + 1];
        const float ta0 = (ya - c0) * g0;
        const float ta1 = (ya - c1) * g1;
        const float tb0 = (yb - c0) * g0;
        const float tb1 = (yb - c1) * g1;
        v2f pa, pb;
        pa.x = __expf(-(ta0 * ta0));
        pa.y = __expf(-(ta1 * ta1));
        pb.x = __expf(-(tb0 * tb0));
        pb.y = __expf(-(tb1 * tb1));
        acca = __builtin_amdgcn_wmma_f32_16x16x4_f32(
            false, a, false, pa, (short)0, acca, false, false);
        accb = __builtin_amdgcn_wmma_f32_16x16x4_f32(
            false, a, false, pb, (short)0, accb, false, false);
    }
    ra = acca[0];
    rb = accb[0];
}

__device__ __forceinline__ float signf_ref(float y) {
    return (y > 0.f) ? 1.f : ((y < 0.f) ? -1.f : 0.f);
}

__device__ __forceinline__ float acc_fun(float u, float y, float rbf) {
    return (u - y * 214.9261f - 19.3607f * signf_ref(y) - rbf + 3.2902f) * INV_95_45;
}

template <int NCH>
__global__ __launch_bounds__(256)
void rk4_rbf_wmma_kernel(const float* __restrict__ inputs,
                         const float* __restrict__ states,
                         const float* __restrict__ centers,
                         const float* __restrict__ gammas,
                         const float* __restrict__ weights,
                         float* __restrict__ out,
                         int B, int K) {
    __shared__ float sC[128];
    __shared__ float sG[128];
    __shared__ float sW[128];
    for (int i = threadIdx.x; i < K; i += blockDim.x) {
        sC[i] = centers[i];
        sG[i] = gammas[i];
        sW[i] = weights[i];
    }
    __syncthreads();

    const int lane  = threadIdx.x & 31;
    const int n     = lane & 15;      // element slot (WMMA N index)
    const int hv    = lane >> 4;      // lane half -> K-pair per WMMA layout
    const int gwave = blockIdx.x * (blockDim.x >> 5) + (threadIdx.x >> 5);

    int elem = gwave * 16 + n;
    const bool valid = (elem < B);
    if (!valid) elem = B - 1;         // clamp: EXEC must stay all-ones for WMMA

    const float u  = inputs[elem];
    const float y1 = states[2 * elem + 1];
    const int  k0     = hv * 2;
    const int  nch_rt = K >> 2;

    // Eval 1 at y1 (yddoti == k1)
    const float r1    = rbf_wmma<NCH>(y1, k0, sC, sG, sW, nch_rt);
    const float acc_a = acc_fun(u, y1, r1);
    const float dy1   = acc_a * RK_DT;
    const float yb    = y1 + 0.5f * dy1;   // k2 == k3 state
    const float yc    = y1 + dy1;          // k4 state

    // Evals 2+3 fused: independent accumulators, shared table loads
    float rb, rc;
    rbf_wmma_dual<NCH>(yb, yc, k0, sC, sG, sW, nch_rt, rb, rc);
    const float acc_b = acc_fun(u, yb, rb);
    const float acc_c = acc_fun(u, yc, rc);

    // st_n = DT/6 * (k1 + 2*k2 + 2*k3 + k4)
    const float st0 = (RK_DT / 6.f) * (y1    + 2.f * yb    + 2.f * yb    + yc);
    const float st1 = (RK_DT / 6.f) * (acc_a + 2.f * acc_b + 2.f * acc_b + acc_c);

    if (valid) {
        out[2 * elem + hv] = hv ? st1 : st0;   // one store per lane
    }
}

extern "C" void kernel_launch(void* const* d_in, const int* in_sizes, int n_in,
                              void* d_out, int out_size, void* d_ws, size_t ws_size,
                              hipStream_t stream) {
    const float* inputs  = (const float*)d_in[0];
    const float* states  = (const float*)d_in[1];
    const float* centers = (const float*)d_in[2];
    const float* gammas  = (const float*)d_in[3];
    const float* weights = (const float*)d_in[4];
    float* out = (float*)d_out;

    const int B = in_sizes[0];   // 1048576
    const int K = in_sizes[2];   // 128

    const int threads = 256;                    // 8 waves -> 128 elements/block
    const int elems_per_block = (threads / 32) * 16;
    const int grid = (B + elems_per_block - 1) / elems_per_block;

    if (K == 128) {
        rk4_rbf_wmma_kernel<32><<<grid, threads, 0, stream>>>(
            inputs, states, centers, gammas, weights, out, B, K);
    } else {
        rk4_rbf_wmma_kernel<0><<<grid, threads, 0, stream>>>(
            inputs, states, centers, gammas, weights, out, B, K);
    }
}